// BottomUpAttention_67190468378730
// MI455X (gfx1250) — compile-verified
//
#include <hip/hip_runtime.h>
#include <math.h>

// Problem constants (match reference)
#define BN 8
#define NN 300
#define RN 36
#define CN 256
#define OUTG 14            // OUT*SR = 7*2
#define NSAMP (OUTG*OUTG)  // 196
#define NTAP (NSAMP*4)     // 784
#define IOU_THR 0.7f
#define CONF_THR 0.3f

#define NMS_THREADS 320
#define ROI_THREADS 256

typedef __attribute__((ext_vector_type(2))) float v2f;
typedef __attribute__((ext_vector_type(8))) float v8f;

// ---------------------------------------------------------------------------
// Kernel 1: per-batch NMS + level assignment. One block per batch.
// ---------------------------------------------------------------------------
__global__ __launch_bounds__(NMS_THREADS)
void nms_kernel(const float* __restrict__ boxes,   // (B,N,4)
                const float* __restrict__ scores,  // (B,N)
                float* __restrict__ selBoxes,      // (B,R,4)
                int*   __restrict__ selMeta)       // (B,R): lv in [0,3] or -1
{
    const int b   = blockIdx.x;
    const int tid = threadIdx.x;
    const float NEG_INF = -__builtin_inff();

    __shared__ float ss[NN];
    __shared__ int   ord[NN];
    __shared__ float bxs[NN][4];
    __shared__ float area[NN];
    __shared__ int   keep[NN];

    // masked scores
    for (int i = tid; i < NN; i += NMS_THREADS) {
        float s = scores[b * NN + i];
        ss[i] = (s > CONF_THR) ? s : NEG_INF;
    }
    __syncthreads();

    // stable descending rank (matches stable jnp.argsort(-s))
    for (int i = tid; i < NN; i += NMS_THREADS) {
        float si = ss[i];
        int rank = 0;
        for (int j = 0; j < NN; ++j) {
            float sj = ss[j];
            rank += (int)((sj > si) || ((sj == si) && (j < i)));
        }
        ord[rank] = i;
    }
    __syncthreads();

    // gather sorted boxes, areas, initial keep
    for (int i = tid; i < NN; i += NMS_THREADS) {
        int oi = ord[i];
        const float* bp = boxes + (size_t)(b * NN + oi) * 4;
        float x1 = bp[0], y1 = bp[1], x2 = bp[2], y2 = bp[3];
        bxs[i][0] = x1; bxs[i][1] = y1; bxs[i][2] = x2; bxs[i][3] = y2;
        area[i] = (x2 - x1) * (y2 - y1);
        keep[i] = (ss[oi] > NEG_INF) ? 1 : 0;
    }
    __syncthreads();

    // greedy suppression, exactly mirroring the reference fori_loop
    for (int i = 0; i < NN - 1; ++i) {
        if (keep[i]) {
            float ax1 = bxs[i][0], ay1 = bxs[i][1];
            float ax2 = bxs[i][2], ay2 = bxs[i][3];
            float ai  = area[i];
            for (int j = i + 1 + tid; j < NN; j += NMS_THREADS) {
                float lx = fmaxf(ax1, bxs[j][0]);
                float ly = fmaxf(ay1, bxs[j][1]);
                float rx = fminf(ax2, bxs[j][2]);
                float ry = fminf(ay2, bxs[j][3]);
                float w = fmaxf(rx - lx, 0.0f);
                float h = fmaxf(ry - ly, 0.0f);
                float inter = w * h;
                float iou = inter / (ai + area[j] - inter + 1e-9f);
                if (iou > IOU_THR) keep[j] = 0;
            }
        }
        __syncthreads();
    }

    // stable kept-first selection of first R; compute level per ROI
    if (tid == 0) {
        int pos = 0;
        for (int pass = 0; pass < 2 && pos < RN; ++pass) {
            for (int i = 0; i < NN && pos < RN; ++i) {
                int k = keep[i];
                if ((pass == 0 && k) || (pass == 1 && !k)) {
                    float x1 = bxs[i][0], y1 = bxs[i][1];
                    float x2 = bxs[i][2], y2 = bxs[i][3];
                    float* sb = selBoxes + (size_t)(b * RN + pos) * 4;
                    sb[0] = x1; sb[1] = y1; sb[2] = x2; sb[3] = y2;
                    int meta = -1;
                    if (k) {
                        float dx = x2 - x1, dy = y2 - y1;
                        float sz = sqrtf(fmaxf(dx * dx + dy * dy, 1e-12f));
                        float t = floorf(4.0f + log2f(sz * (4.0f / 224.0f)));
                        t = fminf(fmaxf(t, 2.0f), 5.0f);
                        meta = (int)t - 2;   // 0..3
                    }
                    selMeta[b * RN + pos] = meta;
                    ++pos;
                }
            }
        }
    }
}

// ---------------------------------------------------------------------------
// Kernel 2: ROI-align pooling. One block (8 wave32) per (b, r) ROI.
// WMMA f32 16x16x4 used as a K-reduction engine: A = value*weight, B = 1.
// Gathers use a wave-uniform scalar base + 32-bit per-lane element index so
// the backend can emit global_load saddr + voffset (GVS, scale_offset).
// ---------------------------------------------------------------------------
__global__ __launch_bounds__(ROI_THREADS)
void roi_pool_kernel(const float* __restrict__ f0,
                     const float* __restrict__ f1,
                     const float* __restrict__ f2,
                     const float* __restrict__ f3,
                     const float* __restrict__ selBoxes,
                     const int*   __restrict__ selMeta,
                     float* __restrict__ out)     // (B,R,C)
{
    const int blk = blockIdx.x;
    const int b   = blk / RN;
    const int tid = threadIdx.x;

    float* outp = out + (size_t)blk * CN;

    const int meta = selMeta[blk];
    if (meta < 0) {                 // invalid ROI -> zeros
        outp[tid] = 0.0f;
        return;
    }

    const int lv = meta;
    const int H  = 200 >> lv;       // 200,100,50,25
    const int HW = H * H;
    const float Hf = (float)H;
    // Ternary select keeps the global address space (avoids FLAT loads).
    const float* __restrict__ feat =
        (lv == 0) ? f0 : (lv == 1) ? f1 : (lv == 2) ? f2 : f3;
    // Wave-uniform base for this batch; per-lane index stays 32-bit
    // (max (255*40000 + 39999) ~= 10.2M elements, well inside i32).
    const float* __restrict__ fb = feat + (size_t)b * CN * HW;

    __shared__ __align__(16) int   offs[NTAP];
    __shared__ __align__(16) float wts[NTAP];

    // per-sample geometry (channel-independent)
    if (tid < NSAMP) {
        const float* sb = selBoxes + (size_t)blk * 4;
        float x1 = sb[0], y1 = sb[1], x2 = sb[2], y2 = sb[3];
        float rw = fmaxf(x2 - x1, 1.0f);
        float rh = fmaxf(y2 - y1, 1.0f);
        int gy = tid / OUTG, gx = tid % OUTG;
        float gyv = ((float)gy + 0.5f) * (1.0f / OUTG);
        float gxv = ((float)gx + 0.5f) * (1.0f / OUTG);
        float y = y1 + rh * gyv;
        float x = x1 + rw * gxv;
        bool oob = (y < -1.0f) || (y > Hf) || (x < -1.0f) || (x > Hf);
        float yc = fminf(fmaxf(y, 0.0f), Hf - 1.0f);
        float xc = fminf(fmaxf(x, 0.0f), Hf - 1.0f);
        int y0 = (int)floorf(yc);
        int x0 = (int)floorf(xc);
        int y1i = min(y0 + 1, H - 1);
        int x1i = min(x0 + 1, H - 1);
        float ly = yc - (float)y0;
        float lx = xc - (float)x0;
        float sc = oob ? 0.0f : (1.0f / (float)NSAMP);
        int t = tid * 4;
        offs[t + 0] = y0  * H + x0;
        offs[t + 1] = y0  * H + x1i;
        offs[t + 2] = y1i * H + x0;
        offs[t + 3] = y1i * H + x1i;
        wts[t + 0] = (1.0f - ly) * (1.0f - lx) * sc;
        wts[t + 1] = (1.0f - ly) * lx * sc;
        wts[t + 2] = ly * (1.0f - lx) * sc;
        wts[t + 3] = ly * lx * sc;
    }
    __syncthreads();

    // WMMA reduction: wave handles 2 chunks of 16 channels.
    // A 16x4 f32 layout: lanes 0-15 carry K={0,1} in v0/v1, lanes 16-31 K={2,3}.
    // B = all ones -> every D column equals the per-row K-sum (layout-robust).
    const int wave = tid >> 5;
    const int lane = tid & 31;
    const int half = lane >> 4;     // 0: K 0,1 ; 1: K 2,3
    const int m    = lane & 15;     // row (channel within chunk)
    const v2f Bones = { 1.0f, 1.0f };

    for (int chunk = 0; chunk < 2; ++chunk) {
        const int c0 = wave * 32 + chunk * 16;
        const int laneBase = (c0 + m) * HW;    // per-lane 32-bit channel base
        v8f acc = { 0.0f, 0.0f, 0.0f, 0.0f, 0.0f, 0.0f, 0.0f, 0.0f };

        #pragma unroll 4
        for (int k = 0; k < NTAP; k += 4) {
            int t = k + (half << 1);           // even -> 8B aligned pair loads
            int2   o2 = *reinterpret_cast<const int2*>(&offs[t]);
            float2 w2 = *reinterpret_cast<const float2*>(&wts[t]);
            int idx0 = laneBase + o2.x;
            int idx1 = laneBase + o2.y;
            float a0 = fb[idx0] * w2.x;
            float a1 = fb[idx1] * w2.y;
            v2f A = { a0, a1 };
            acc = __builtin_amdgcn_wmma_f32_16x16x4_f32(
                false, A, false, Bones, (short)0, acc, false, false);
        }

        // D layout: VGPR v -> M=v (lanes 0-15), M=v+8 (lanes 16-31); all N equal.
        // lane 0 holds rows 0-7, lane 16 holds rows 8-15; write as 2x b128.
        if (lane == 0 || lane == 16) {
            float* dst = outp + c0 + (half << 3);
            float4 lo = make_float4(acc[0], acc[1], acc[2], acc[3]);
            float4 hi = make_float4(acc[4], acc[5], acc[6], acc[7]);
            *reinterpret_cast<float4*>(dst)     = lo;
            *reinterpret_cast<float4*>(dst + 4) = hi;
        }
    }
}

// ---------------------------------------------------------------------------
extern "C" void kernel_launch(void* const* d_in, const int* in_sizes, int n_in,
                              void* d_out, int out_size, void* d_ws, size_t ws_size,
                              hipStream_t stream) {
    (void)in_sizes; (void)n_in; (void)out_size; (void)ws_size;
    const float* boxes  = (const float*)d_in[0];
    const float* scores = (const float*)d_in[1];
    const float* f0     = (const float*)d_in[2];
    const float* f1     = (const float*)d_in[3];
    const float* f2     = (const float*)d_in[4];
    const float* f3     = (const float*)d_in[5];

    float* selBoxes = (float*)d_ws;
    int*   selMeta  = (int*)((char*)d_ws + (size_t)BN * RN * 4 * sizeof(float));

    nms_kernel<<<BN, NMS_THREADS, 0, stream>>>(boxes, scores, selBoxes, selMeta);
    roi_pool_kernel<<<BN * RN, ROI_THREADS, 0, stream>>>(
        f0, f1, f2, f3, selBoxes, selMeta, (float*)d_out);
}